// DynamicMHCAdapter_24137716203609
// MI455X (gfx1250) — compile-verified
//
#include <hip/hip_runtime.h>

typedef __attribute__((ext_vector_type(2))) float v2f;
typedef __attribute__((ext_vector_type(8))) float v8f;

#define NCDIM 8192      // n_heads * C
#define OUTDIM 24       // n*n + 2n
#define EPSV 1e-20f

// One wave computes proj[16 tokens x 24 outs] via V_WMMA_F32_16X16X4_F32,
// plus ||x||^2 on the fly, then a per-token Sinkhorn epilogue on lanes 0-15.
__launch_bounds__(128, 1)
__global__ void mhc_adapter_kernel(const float* __restrict__ x,
                                   const float* __restrict__ W,
                                   const float* __restrict__ bias,
                                   const float* __restrict__ pAlphaRes,
                                   const float* __restrict__ pAlphaPre,
                                   const float* __restrict__ pAlphaPost,
                                   float* __restrict__ out,
                                   int tokens) {
    __shared__ float smem[4][16][25];   // [wave][token-in-tile][out col], padded stride

    const int lane = threadIdx.x & 31;
    const int wave = threadIdx.x >> 5;
    const int li   = lane & 15;         // M (A) / N (B) index within tile
    const int half = lane >> 4;         // selects K pair {0,1} vs {2,3}
    const int tokBase = (blockIdx.x * 4 + wave) * 16;
    if (tokBase >= tokens) return;

    // second B tile covers outputs 16..23; clamp invalid lanes to a valid W row
    const int n1 = (li < 8) ? (16 + li) : 23;

    const float* aPtr  = x + (size_t)(tokBase + li) * NCDIM + (half << 1);
    const float* b0Ptr = W + (size_t)li * NCDIM + (half << 1);
    const float* b1Ptr = W + (size_t)n1 * NCDIM + (half << 1);

    v8f acc0 = {};   // proj cols 0..15
    v8f acc1 = {};   // proj cols 16..23 (cols 8..15 garbage, never read)
    float sumsq = 0.0f;

    for (int k = 0; k < NCDIM; k += 16) {
#pragma unroll
        for (int j = 0; j < 4; ++j) {
            const int kk = k + 4 * j;
            v2f a  = *(const v2f*)(aPtr  + kk);
            v2f b0 = *(const v2f*)(b0Ptr + kk);
            v2f b1 = *(const v2f*)(b1Ptr + kk);
            sumsq += a.x * a.x + a.y * a.y;
            // D = A(16x4, f32) x B(4x16, f32) + C  (8 args: neg_a,A,neg_b,B,c_mod,C,reuse_a,reuse_b)
            acc0 = __builtin_amdgcn_wmma_f32_16x16x4_f32(false, a, false, b0,
                                                         (short)0, acc0, false, false);
            acc1 = __builtin_amdgcn_wmma_f32_16x16x4_f32(false, a, false, b1,
                                                         (short)0, acc1, false, false);
        }
    }

    // lanes 0-15 hold K%4 in {0,1}, lanes 16-31 hold {2,3} for the same token: fold halves
    float tot = sumsq + __shfl_xor(sumsq, 16, 32);
    float r   = sqrtf(tot * (1.0f / NCDIM)) + EPSV;   // == ||x||/sqrt(nC) + eps

    // Scatter C/D fragments to LDS, row-major per token.
    // C/D layout: VGPR v holds (M = v + 8*half, N = li).
#pragma unroll
    for (int v = 0; v < 8; ++v)
        smem[wave][v + 8 * half][li] = acc0[v];
    if (li < 8) {
#pragma unroll
        for (int v = 0; v < 8; ++v)
            smem[wave][v + 8 * half][16 + li] = acc1[v];
    }
    __syncthreads();

    if (lane < 16) {
        const float aRes  = pAlphaRes[0];
        const float aPre  = pAlphaPre[0];
        const float aPost = pAlphaPost[0];
        const float inv_r = 1.0f / r;       // lane t holds r for token tokBase+t
        const size_t tok  = (size_t)tokBase + lane;

        // l_res -> 4x4 Sinkhorn in registers
        float P[16];
#pragma unroll
        for (int i = 0; i < 16; ++i)
            P[i] = aRes * smem[wave][lane][i] * inv_r + bias[i];

        for (int it = 0; it < 20; ++it) {
#pragma unroll
            for (int rI = 0; rI < 4; ++rI) {            // row logsumexp
                float m = fmaxf(fmaxf(P[4*rI], P[4*rI+1]), fmaxf(P[4*rI+2], P[4*rI+3]));
                float s = expf(P[4*rI]   - m) + expf(P[4*rI+1] - m)
                        + expf(P[4*rI+2] - m) + expf(P[4*rI+3] - m);
                float lse = m + logf(s);
#pragma unroll
                for (int c = 0; c < 4; ++c) P[4*rI + c] -= lse;
            }
#pragma unroll
            for (int cI = 0; cI < 4; ++cI) {            // column logsumexp
                float m = fmaxf(fmaxf(P[cI], P[cI+4]), fmaxf(P[cI+8], P[cI+12]));
                float s = expf(P[cI]   - m) + expf(P[cI+4]  - m)
                        + expf(P[cI+8] - m) + expf(P[cI+12] - m);
                float lse = m + logf(s);
#pragma unroll
                for (int rr = 0; rr < 4; ++rr) P[cI + 4*rr] -= lse;
            }
        }

        // outputs concatenated flat: H_res [tokens*16], H_pre [tokens*4], H_post [tokens*4]
        float* out_res  = out;
        float* out_pre  = out + (size_t)tokens * 16;
        float* out_post = out + (size_t)tokens * 20;
#pragma unroll
        for (int i = 0; i < 16; ++i)
            out_res[tok * 16 + i] = expf(P[i]);
#pragma unroll
        for (int i = 0; i < 4; ++i) {
            float lp = aPre  * smem[wave][lane][16 + i] * inv_r + bias[16 + i];
            out_pre[tok * 4 + i]  = 1.0f / (1.0f + expf(-lp));
            float lq = aPost * smem[wave][lane][20 + i] * inv_r + bias[20 + i];
            out_post[tok * 4 + i] = 2.0f / (1.0f + expf(-lq));
        }
    }
}

extern "C" void kernel_launch(void* const* d_in, const int* in_sizes, int n_in,
                              void* d_out, int out_size, void* d_ws, size_t ws_size,
                              hipStream_t stream) {
    const float* x      = (const float*)d_in[0];
    const float* W      = (const float*)d_in[1];
    const float* bias   = (const float*)d_in[2];
    const float* aRes   = (const float*)d_in[3];
    const float* aPre   = (const float*)d_in[4];
    const float* aPost  = (const float*)d_in[5];
    float* out          = (float*)d_out;

    const int tokens = in_sizes[0] / NCDIM;       // B*L = 8192
    const int blocks = (tokens + 63) / 64;        // 64 tokens per 128-thread block
    mhc_adapter_kernel<<<blocks, 128, 0, stream>>>(x, W, bias, aRes, aPre, aPost,
                                                   out, tokens);
}